// GroupSort_48215302865068
// MI455X (gfx1250) — compile-verified
//
#include <hip/hip_runtime.h>
#include <stdint.h>

// GroupSort (k=4): sort each contiguous group of 4 floats along the last dim.
// B=16384, D=4096 -> 16,777,216 float4 groups. Pure streaming: 512MB @ 23.3TB/s
// => ~22us floor. Data path: 4-deep async global->LDS pipeline (CDNA5 async-copy
// path, ~1.5KB in flight per wave for HBM saturation), 5-comparator min/max
// network in registers, non-temporal b128 stores.

#define TPB 256
#define NBUF 4              // async pipeline depth (power of two)
#define GROUPS_PER_THREAD 16

#define GLB_AS __attribute__((address_space(1)))
#define LDS_AS __attribute__((address_space(3)))

typedef float v4f __attribute__((ext_vector_type(4)));
typedef int   v4i __attribute__((ext_vector_type(4)));

#if defined(__AMDGCN__) && defined(__gfx1250__) && \
    __has_builtin(__builtin_amdgcn_global_load_async_to_lds_b128)
#define USE_ASYNC 1
#else
#define USE_ASYNC 0
#endif

#if USE_ASYNC
template <int N>
__device__ __forceinline__ void wait_async() {
#if __has_builtin(__builtin_amdgcn_s_wait_asynccnt)
  __builtin_amdgcn_s_wait_asynccnt(N);
#else
  asm volatile("s_wait_asynccnt %0" ::"i"(N) : "memory");
#endif
}
#endif

// 5-comparator sorting network for 4 elements (ascending):
// (0,1)(2,3)(0,2)(1,3)(1,2) -> 10 v_min/v_max ops per 16 bytes.
__device__ __forceinline__ v4f sort4(v4f v) {
  float a = v.x, b = v.y, c = v.z, d = v.w, t;
  t = fminf(a, b); b = fmaxf(a, b); a = t;
  t = fminf(c, d); d = fmaxf(c, d); c = t;
  t = fminf(a, c); c = fmaxf(a, c); a = t;
  t = fminf(b, d); d = fmaxf(b, d); b = t;
  t = fminf(b, c); c = fmaxf(b, c); b = t;
  v4f r; r.x = a; r.y = b; r.z = c; r.w = d;
  return r;
}

__global__ __launch_bounds__(TPB) void groupsort4_kernel(
    const v4f* __restrict__ x, v4f* __restrict__ y,
    unsigned nGroups, unsigned gsize /* gridDim.x*TPB */,
    unsigned iters /* nGroups / gsize */) {
#if USE_ASYNC
  // 4-stage staging: [stage][thread] -> 4*256*16B = 16KB LDS per block.
  __shared__ v4f stage[NBUF][TPB];
  const unsigned tid = threadIdx.x;
  const unsigned i0 = blockIdx.x * TPB + tid;
  GLB_AS v4i* gx = (GLB_AS v4i*)x;

  // Prologue: fill up to NBUF-1 stages.
#pragma unroll
  for (unsigned p = 0; p < NBUF - 1; ++p) {
    if (p < iters)
      __builtin_amdgcn_global_load_async_to_lds_b128(
          gx + (size_t)i0 + (size_t)p * gsize,
          (LDS_AS v4i*)&stage[p][tid], 0, 0);
  }

  unsigned i = i0;
  for (unsigned it = 0; it < iters; ++it) {
    if (it + NBUF <= iters) {
      // Steady state: keep NBUF-1 loads in flight.
      __builtin_amdgcn_global_load_async_to_lds_b128(
          gx + (size_t)i + (size_t)(NBUF - 1) * gsize,
          (LDS_AS v4i*)&stage[(it + NBUF - 1) & (NBUF - 1)][tid], 0, 0);
      wait_async<NBUF - 1>();  // in-order completion: stage `it` is ready
    } else {
      // Drain: exactly (iters-1-it) loads still outstanding are allowed.
      const unsigned rem = iters - 1u - it;
      if (rem >= 2u)      wait_async<2>();
      else if (rem == 1u) wait_async<1>();
      else                wait_async<0>();
    }
    asm volatile("" ::: "memory");       // keep the ds_load below the wait
    v4f v = stage[it & (NBUF - 1)][tid]; // ds_load_b128, conflict-free
    v = sort4(v);
    __builtin_nontemporal_store(v, &y[i]);  // write-once output: bypass cache
    i += gsize;
  }
  // Ragged tail (empty for the 2^24-group problem, kept for generality).
  if (i < nGroups) {
    v4f v = sort4(__builtin_nontemporal_load(&x[i]));
    __builtin_nontemporal_store(v, &y[i]);
  }
#else
  (void)iters;
  const unsigned gs = gsize;
  for (unsigned i = blockIdx.x * blockDim.x + threadIdx.x; i < nGroups; i += gs) {
    v4f v = sort4(__builtin_nontemporal_load(&x[i]));
    __builtin_nontemporal_store(v, &y[i]);
  }
#endif
}

extern "C" void kernel_launch(void* const* d_in, const int* in_sizes, int n_in,
                              void* d_out, int out_size, void* d_ws, size_t ws_size,
                              hipStream_t stream) {
  (void)n_in; (void)out_size; (void)d_ws; (void)ws_size;
  const v4f* x = (const v4f*)d_in[0];
  // d_in[1] is k (device scalar); the reference fixes k=4 and the sorting
  // network is compiled for k=4.
  v4f* y = (v4f*)d_out;

  const unsigned nGroups = (unsigned)(in_sizes[0] / 4);
  const unsigned perBlock = TPB * GROUPS_PER_THREAD;
  unsigned blocks = (nGroups + perBlock - 1) / perBlock;
  if (blocks == 0) blocks = 1;

  const unsigned gsize = blocks * TPB;
  const unsigned iters = nGroups / gsize;  // uniform trip count (host-side div)

  groupsort4_kernel<<<blocks, TPB, 0, stream>>>(x, y, nGroups, gsize, iters);
}